// EDModelQP_69415261438711
// MI455X (gfx1250) — compile-verified
//
#include <hip/hip_runtime.h>
#include <math.h>

// Problem constants (match reference)
constexpr int kNZ = 480;   // decision vars
constexpr int kM  = 960;   // inequality rows
constexpr int kNE = 200;   // equality rows
constexpr int kB  = 16;    // batch
constexpr int kIters = 25;
constexpr float kSigma = 0.1f;
constexpr float kBigF = 3.0e38f;

typedef __attribute__((ext_vector_type(2))) float v2f;
typedef __attribute__((ext_vector_type(4))) float v4f;
typedef __attribute__((ext_vector_type(8))) float v8f;

// ---------------- workspace layout (floats) ----------------
// Global section (shared across batches): Gt = G^T, 480x960
constexpr int GT_SZ   = kNZ * kM;
// Per-batch section:
constexpr int OFF_Z    = 0;
constexpr int OFF_S    = OFF_Z   + kNZ;
constexpr int OFF_LAM  = OFF_S   + kM;
constexpr int OFF_NU   = OFF_LAM + kM;
constexpr int OFF_W    = OFF_NU  + kNE;
constexpr int OFF_RP   = OFF_W   + kM;
constexpr int OFF_RC   = OFF_RP  + kM;
constexpr int OFF_RE   = OFF_RC  + kM;
constexpr int OFF_R1   = OFF_RE  + kNE;
constexpr int OFF_U    = OFF_R1  + kNZ;
constexpr int OFF_DNU  = OFF_U   + kNZ;
constexpr int OFF_MRED = OFF_DNU + kNE;          // 480x480, Cholesky in place
constexpr int OFF_Y    = OFF_MRED + kNZ * kNZ;   // 480x200 row-major: Y[i][j] = (Mred^-1 A^T)[i][j]
constexpr int OFF_SCH  = OFF_Y    + kNZ * kNE;   // 200x200 Schur complement
constexpr int OFF_GWT  = OFF_SCH  + kNE * kNE;   // 480x960 : (diag(W) G)^T
constexpr int WS_PER_B = OFF_GWT  + kNZ * kM;    // ~3.3 MB/batch; total ~55 MB

__device__ __forceinline__ float* batch_ws(float* ws, int b) {
  return ws + GT_SZ + (size_t)b * WS_PER_B;
}

// ---------------- Gt = G^T (once per launch) ----------------
__global__ __launch_bounds__(kM) void k_transpose(const float* __restrict__ G,
                                                  float* __restrict__ ws) {
  const int i = blockIdx.x, k = threadIdx.x;          // i<480, k<960
  ws[(size_t)i * kM + k] = G[(size_t)k * kNZ + i];    // coalesced store
}

// ---------------- init: z=0, s=1, lam=1, nu=0 ----------------
__global__ __launch_bounds__(kM) void k_init(float* __restrict__ ws) {
  const int b = blockIdx.x, t = threadIdx.x;
  float* w = batch_ws(ws, b);
  if (t < kNZ) w[OFF_Z + t] = 0.f;
  if (t < kM)  { w[OFF_S + t] = 1.f; w[OFF_LAM + t] = 1.f; }
  if (t < kNE) w[OFF_NU + t] = 0.f;
}

// ---------------- residuals: W, mu, rp, rc, re, r1 ----------------
__global__ __launch_bounds__(kM) void k_residuals(
    const float* __restrict__ Q, const float* __restrict__ p,
    const float* __restrict__ G, const float* __restrict__ h,
    const float* __restrict__ A, const float* __restrict__ bvec,
    float* __restrict__ ws) {
  const int b = blockIdx.x, t = threadIdx.x;
  float* w = batch_ws(ws, b);
  float* z   = w + OFF_Z;   float* s  = w + OFF_S;
  float* lam = w + OFF_LAM; float* nu = w + OFF_NU;
  float* W   = w + OFF_W;   float* rp = w + OFF_RP;
  float* rc  = w + OFF_RC;  float* re = w + OFF_RE;
  float* r1  = w + OFF_R1;
  const float* pb = p + (size_t)b * kNZ;
  const float* hb = h + (size_t)b * kM;
  const float* bb = bvec + (size_t)b * kNE;

  __shared__ float red[kM];
  __shared__ float tbuf[kM];

  float lv = 0.f, sv = 1.f;
  if (t < kM) { lv = lam[t]; sv = s[t]; W[t] = lv / sv; red[t] = lv * sv; }
  __syncthreads();
  for (int off = 512; off > 0; off >>= 1) {
    if (t < off && t + off < kM) red[t] += red[t + off];
    __syncthreads();
  }
  const float mu = red[0] / (float)kM;

  if (t < kM) {
    float acc = 0.f;
    const float* gr = G + (size_t)t * kNZ;
    for (int k = 0; k < kNZ; ++k) acc += gr[k] * z[k];
    const float rpv = acc + sv - hb[t];
    const float rcv = lv * sv - kSigma * mu;
    rp[t] = rpv; rc[t] = rcv;
    tbuf[t] = (lv / sv) * rpv - rcv / sv;   // W*rp - rc/s
  }
  if (t < kNE) {
    float acc = 0.f;
    const float* ar = A + (size_t)t * kNZ;
    for (int k = 0; k < kNZ; ++k) acc += ar[k] * z[k];
    re[t] = acc - bb[t];
  }
  __syncthreads();
  if (t < kNZ) {
    float rd = pb[t];
    const float* qr = Q + (size_t)t * kNZ;
    for (int k = 0; k < kNZ; ++k) rd += qr[k] * z[k];          // Q z
    float g1 = 0.f, g2 = 0.f;
    for (int k = 0; k < kM; ++k) {
      const float gk = G[(size_t)k * kNZ + t];                 // coalesced across t
      g1 += gk * lam[k];                                       // G^T lam
      g2 += gk * tbuf[k];                                      // G^T (W rp - rc/s)
    }
    for (int k = 0; k < kNE; ++k) rd += A[(size_t)k * kNZ + t] * nu[k]; // A^T nu
    r1[t] = -(rd + g1 + g2);
  }
}

// ---------------- GWt = (diag(W) G)^T  (per batch, per iter) ----------------
__global__ __launch_bounds__(kM) void k_scale_GW(float* __restrict__ ws) {
  const int i = blockIdx.x, b = blockIdx.y, k = threadIdx.x;
  const float* Gt = ws;                                // 480x960
  float* w = batch_ws(ws, b);
  w[OFF_GWT + (size_t)i * kM + k] = Gt[(size_t)i * kM + k] * w[OFF_W + k];
}

// ---------------- Mred = Q + G^T diag(W) G  (f32 WMMA, LDS-staged) ----------------
// 4 waves per WG; each wave one 16x16 tile of a 32x32 block. Panels of
// Gt/GWt staged in LDS via b128 loads; fragments gathered as ds_load_b64.
__global__ __launch_bounds__(128) void k_build_mred(
    const float* __restrict__ Q, float* __restrict__ ws) {
  const int b = blockIdx.z;
  const float* Gt = ws;                                // 480x960 (A operand, K-contig)
  float* w = batch_ws(ws, b);
  const float* GWt = w + OFF_GWT;                      // 480x960 (B operand, K-contig)
  float* Mred = w + OFF_MRED;

  const int t    = threadIdx.x;
  const int lane = t & 31;
  const int wv   = t >> 5;                             // wave 0..3
  const int m    = lane & 15;                          // row/col within tile
  const int kh   = lane >> 4;                          // K-half: K = 2*kh + vgpr
  const int ib = blockIdx.x * 32, jb = blockIdx.y * 32;
  const int il = (wv >> 1) * 16;                       // local tile origin
  const int jl = (wv & 1) * 16;

  constexpr int KC  = 64;                              // K chunk
  constexpr int LDP = KC + 4;                          // padded row stride (16B-aligned)
  __shared__ float As[32 * LDP];
  __shared__ float Bs[32 * LDP];

  const int lr = t >> 4;                               // 0..7 loader row group
  const int lc = (t & 15) * 4;                         // loader col (x4 floats)

  v8f c = {0.f, 0.f, 0.f, 0.f, 0.f, 0.f, 0.f, 0.f};
  for (int kc = 0; kc < kM; kc += KC) {
    __syncthreads();
    const int kpf = (kc + KC < kM) ? (kc + KC) : kc;   // uniform clamped prefetch
    __builtin_prefetch(&Gt[(size_t)(ib + lr) * kM + kpf + lc], 0, 0);
    for (int r = 0; r < 4; ++r) {
      const int row = lr + r * 8;                      // 0..31
      *(v4f*)&As[row * LDP + lc] = *(const v4f*)&Gt [(size_t)(ib + row) * kM + kc + lc];
      *(v4f*)&Bs[row * LDP + lc] = *(const v4f*)&GWt[(size_t)(jb + row) * kM + kc + lc];
    }
    __syncthreads();
    for (int kk = 0; kk < KC; kk += 4) {
      const v2f a  = *(const v2f*)&As[(il + m) * LDP + kk + 2 * kh];  // ds_load_b64
      const v2f bf = *(const v2f*)&Bs[(jl + m) * LDP + kk + 2 * kh];  // ds_load_b64
      c = __builtin_amdgcn_wmma_f32_16x16x4_f32(false, a, false, bf,
                                                (short)0, c, false, false);
    }
  }
  // C layout: vgpr v, lanes 0-15 -> M=v, lanes 16-31 -> M=v+8; N = lane&15
  for (int v = 0; v < 8; ++v) {
    const int row = ib + il + v + 8 * kh;
    const int col = jb + jl + m;
    Mred[(size_t)row * kNZ + col] = c[v] + Q[(size_t)row * kNZ + col];
  }
}

// ---------------- Cholesky of Mred (480x480, in place) ----------------
__global__ __launch_bounds__(kNZ) void k_chol_mred(float* __restrict__ ws) {
  const int b = blockIdx.x, t = threadIdx.x;
  float* Mm = batch_ws(ws, b) + OFF_MRED;
  for (int j = 0; j < kNZ; ++j) {
    __syncthreads();
    if (t == 0) Mm[(size_t)j * kNZ + j] = sqrtf(Mm[(size_t)j * kNZ + j]);
    __syncthreads();
    const float dinv = 1.f / Mm[(size_t)j * kNZ + j];
    const int i = j + 1 + t;                    // each thread owns one row
    if (i < kNZ) Mm[(size_t)i * kNZ + j] *= dinv;
    __syncthreads();
    if (i < kNZ) {
      const float lij = Mm[(size_t)i * kNZ + j];
      for (int k = j + 1; k <= i; ++k)
        Mm[(size_t)i * kNZ + k] -= lij * Mm[(size_t)k * kNZ + j];
    }
  }
}

// ---------------- Y = Mred^-1 A^T, all 200 RHS via rank-1 column sweep --------
// Y stored 480x200 row-major; trailing block is L2-resident (384 KB/batch).
__global__ __launch_bounds__(1024) void k_solve_Y(const float* __restrict__ A,
                                                  float* __restrict__ ws) {
  const int b = blockIdx.x, t = threadIdx.x;
  float* w = batch_ws(ws, b);
  const float* L = w + OFF_MRED;
  float* Yn = w + OFF_Y;                               // 480 x 200
  __shared__ float yrow[kNE];

  for (int idx = t; idx < kNZ * kNE; idx += 1024) {    // Yn = A^T
    const int i = idx / kNE, j = idx - i * kNE;
    Yn[idx] = A[(size_t)j * kNZ + i];
  }
  __syncthreads();
  for (int i = 0; i < kNZ; ++i) {                      // forward: L Y' = A^T
    const float dinv = 1.f / L[(size_t)i * kNZ + i];
    if (t < kNE) {
      const float yv = Yn[(size_t)i * kNE + t] * dinv;
      yrow[t] = yv; Yn[(size_t)i * kNE + t] = yv;
    }
    __syncthreads();
    const int n = (kNZ - 1 - i) * kNE;
    for (int idx = t; idx < n; idx += 1024) {
      const int r = idx / kNE, j = idx - r * kNE;
      const int ii = i + 1 + r;
      Yn[(size_t)ii * kNE + j] -= L[(size_t)ii * kNZ + i] * yrow[j];
    }
    __syncthreads();
  }
  for (int i = kNZ - 1; i >= 0; --i) {                 // backward: L^T Y = Y'
    const float dinv = 1.f / L[(size_t)i * kNZ + i];
    if (t < kNE) {
      const float yv = Yn[(size_t)i * kNE + t] * dinv;
      yrow[t] = yv; Yn[(size_t)i * kNE + t] = yv;
    }
    __syncthreads();
    const int n = i * kNE;
    for (int idx = t; idx < n; idx += 1024) {
      const int ii = idx / kNE, j = idx - ii * kNE;
      Yn[(size_t)ii * kNE + j] -= L[(size_t)i * kNZ + ii] * yrow[j];
    }
    __syncthreads();
  }
}

// ---------------- S = A * Y + 1e-8 I  (f32 WMMA, edge-masked) ----------------
__global__ __launch_bounds__(32) void k_build_S(const float* __restrict__ A,
                                                float* __restrict__ ws) {
  const int b = blockIdx.z;
  float* w = batch_ws(ws, b);
  const float* Yn = w + OFF_Y;                         // 480 x 200
  float* S = w + OFF_SCH;
  const int lane = threadIdx.x;
  const int m = lane & 15, kh = lane >> 4;
  const int ib = blockIdx.x * 16, jb = blockIdx.y * 16;
  const int ra = ib + m;      // A row for A-fragment
  const int cb = jb + m;      // Y column for B-fragment (= output column)
  const int rac = (ra < kNE) ? ra : 0;                 // clamp to stay in-bounds

  v8f c = {0.f, 0.f, 0.f, 0.f, 0.f, 0.f, 0.f, 0.f};
  for (int k0 = 0; k0 < kNZ; k0 += 4) {
    const int ka = k0 + 2 * kh;
    v2f a = *(const v2f*)&A[(size_t)rac * kNZ + ka];   // contiguous K-pair
    if (ra >= kNE) { a.x = 0.f; a.y = 0.f; }
    v2f bf;
    bf.x = (cb < kNE) ? Yn[(size_t)(ka + 0) * kNE + cb] : 0.f;
    bf.y = (cb < kNE) ? Yn[(size_t)(ka + 1) * kNE + cb] : 0.f;
    c = __builtin_amdgcn_wmma_f32_16x16x4_f32(false, a, false, bf,
                                              (short)0, c, false, false);
  }
  for (int v = 0; v < 8; ++v) {
    const int row = ib + v + 8 * kh;
    const int col = jb + m;
    if (row < kNE && col < kNE)
      S[(size_t)row * kNE + col] = c[v] + (row == col ? 1e-8f : 0.f);
  }
}

// ---------------- u = Mred^-1 r1 (column-sweep triangular solves) ----------------
__global__ __launch_bounds__(kNZ) void k_solve_u(float* __restrict__ ws) {
  const int b = blockIdx.x, t = threadIdx.x;
  float* w = batch_ws(ws, b);
  const float* L = w + OFF_MRED;
  float* u = w + OFF_U;
  __shared__ float sh;
  float acc = w[OFF_R1 + t];
  for (int j = 0; j < kNZ; ++j) {               // L y = r1
    if (t == j) { sh = acc / L[(size_t)j * kNZ + j]; u[j] = sh; }
    __syncthreads();
    const float uj = sh;
    if (t > j) acc -= L[(size_t)t * kNZ + j] * uj;
    __syncthreads();
  }
  float acc2 = u[t];
  for (int j = kNZ - 1; j >= 0; --j) {          // L^T u = y
    if (t == j) { sh = acc2 / L[(size_t)j * kNZ + j]; u[j] = sh; }
    __syncthreads();
    const float xj = sh;
    if (t < j) acc2 -= L[(size_t)j * kNZ + t] * xj;
    __syncthreads();
  }
}

// ---------------- Schur: chol(S), solve S dnu = A u + re ----------------
__global__ __launch_bounds__(256) void k_schur_solve(const float* __restrict__ A,
                                                     float* __restrict__ ws) {
  const int b = blockIdx.x, t = threadIdx.x;
  float* w = batch_ws(ws, b);
  float* S = w + OFF_SCH;
  const float* u = w + OFF_U;
  const float* re = w + OFF_RE;
  float* dnu = w + OFF_DNU;
  __shared__ float rn[kNE];

  if (t < kNE) {                                // rhs = A u - r2 = A u + re
    float acc = re[t];
    const float* ar = A + (size_t)t * kNZ;
    for (int k = 0; k < kNZ; ++k) acc += ar[k] * u[k];
    rn[t] = acc;
  }
  __syncthreads();
  for (int j = 0; j < kNE; ++j) {               // Cholesky of S
    if (t == 0) S[(size_t)j * kNE + j] = sqrtf(S[(size_t)j * kNE + j]);
    __syncthreads();
    const float dinv = 1.f / S[(size_t)j * kNE + j];
    for (int i = j + 1 + t; i < kNE; i += 256) S[(size_t)i * kNE + j] *= dinv;
    __syncthreads();
    for (int i = j + 1 + t; i < kNE; i += 256) {
      const float lij = S[(size_t)i * kNE + j];
      for (int k = j + 1; k <= i; ++k)
        S[(size_t)i * kNE + k] -= lij * S[(size_t)k * kNE + j];
    }
    __syncthreads();
  }
  for (int j = 0; j < kNE; ++j) {               // forward solve
    if (t == 0) rn[j] /= S[(size_t)j * kNE + j];
    __syncthreads();
    const float wj = rn[j];
    if (t > j && t < kNE) rn[t] -= S[(size_t)t * kNE + j] * wj;
    __syncthreads();
  }
  for (int j = kNE - 1; j >= 0; --j) {          // backward solve
    if (t == 0) rn[j] /= S[(size_t)j * kNE + j];
    __syncthreads();
    const float wj = rn[j];
    if (t < j) rn[t] -= S[(size_t)j * kNE + t] * wj;
    __syncthreads();
  }
  if (t < kNE) dnu[t] = rn[t];
}

// ---------------- dz, dlam, ds, step length, state update ----------------
__global__ __launch_bounds__(kM) void k_finish(const float* __restrict__ G,
                                               float* __restrict__ ws) {
  const int b = blockIdx.x, t = threadIdx.x;
  float* w = batch_ws(ws, b);
  float* z   = w + OFF_Z;   float* s  = w + OFF_S;
  float* lam = w + OFF_LAM; float* nu = w + OFF_NU;
  const float* W  = w + OFF_W;  const float* rp = w + OFF_RP;
  const float* rc = w + OFF_RC; const float* u  = w + OFF_U;
  const float* dnu = w + OFF_DNU; const float* Yn = w + OFF_Y;
  __shared__ float dzs[kNZ];
  __shared__ float red[kM];

  if (t < kNZ) {                                // dz = u - Y dnu (row-contiguous)
    float dv = u[t];
    for (int j = 0; j < kNE; ++j) dv -= Yn[(size_t)t * kNE + j] * dnu[j];
    dzs[t] = dv;
  }
  __syncthreads();

  float dlamv = 0.f, dsv = 0.f, ratio = kBigF;
  if (t < kM) {
    float acc = 0.f;
    const float* gr = G + (size_t)t * kNZ;
    for (int k = 0; k < kNZ; ++k) acc += gr[k] * dzs[k];      // G dz
    const float sv = s[t], lv = lam[t];
    dlamv = W[t] * (acc + rp[t]) - rc[t] / sv;
    dsv   = -(rc[t] + sv * dlamv) / lv;
    const float r1v = (dlamv < 0.f) ? (-lv / dlamv) : kBigF;
    const float r2v = (dsv   < 0.f) ? (-sv / dsv)   : kBigF;
    ratio = fminf(r1v, r2v);
  }
  red[t] = ratio;
  __syncthreads();
  for (int off = 512; off > 0; off >>= 1) {
    if (t < off && t + off < kM) red[t] = fminf(red[t], red[t + off]);
    __syncthreads();
  }
  const float alpha = 0.99f * fminf(1.0f, red[0]);
  if (t < kNZ) z[t] += alpha * dzs[t];
  if (t < kM)  { s[t] += alpha * dsv; lam[t] += alpha * dlamv; }
  if (t < kNE) nu[t] += alpha * dnu[t];
}

// ---------------- write z to output ----------------
__global__ __launch_bounds__(kNZ) void k_writeout(const float* __restrict__ ws,
                                                  float* __restrict__ out) {
  const int b = blockIdx.x, t = threadIdx.x;
  out[(size_t)b * kNZ + t] = ws[GT_SZ + (size_t)b * WS_PER_B + OFF_Z + t];
}

extern "C" void kernel_launch(void* const* d_in, const int* in_sizes, int n_in,
                              void* d_out, int out_size, void* d_ws, size_t ws_size,
                              hipStream_t stream) {
  (void)in_sizes; (void)n_in; (void)out_size; (void)ws_size;
  const float* Q    = (const float*)d_in[0];
  const float* p    = (const float*)d_in[1];
  const float* G    = (const float*)d_in[2];
  const float* h    = (const float*)d_in[3];
  const float* A    = (const float*)d_in[4];
  const float* bvec = (const float*)d_in[5];
  float* out = (float*)d_out;
  float* ws  = (float*)d_ws;

  k_transpose<<<kNZ, kM, 0, stream>>>(G, ws);
  k_init<<<kB, kM, 0, stream>>>(ws);
  for (int it = 0; it < kIters; ++it) {
    k_residuals<<<kB, kM, 0, stream>>>(Q, p, G, h, A, bvec, ws);
    k_scale_GW<<<dim3(kNZ, kB), kM, 0, stream>>>(ws);
    k_build_mred<<<dim3(kNZ / 32, kNZ / 32, kB), 128, 0, stream>>>(Q, ws);
    k_chol_mred<<<kB, kNZ, 0, stream>>>(ws);
    k_solve_Y<<<kB, 1024, 0, stream>>>(A, ws);
    k_build_S<<<dim3(13, 13, kB), 32, 0, stream>>>(A, ws);
    k_solve_u<<<kB, kNZ, 0, stream>>>(ws);
    k_schur_solve<<<kB, 256, 0, stream>>>(A, ws);
    k_finish<<<kB, kM, 0, stream>>>(G, ws);
  }
  k_writeout<<<kB, kNZ, 0, stream>>>(ws, out);
}